// CNFSolver_25735444038418
// MI455X (gfx1250) — compile-verified
//
#include <hip/hip_runtime.h>

// CNF reversible backward solve, persistent-tile kernel for gfx1250.
//
// B=256 rows are independent -> 16 workgroups, each owning a 16-row tile for
// all 64 Euler steps. Weights + state live in LDS; GEMMs use
// V_WMMA_F32_16X16X4_F32 (exact fp32). GEMM2 is fused into GEMM1's N-tile
// loop via a 16x16 LDS staging transpose, so the [256,256] hidden activation
// is never materialized. Trace uses trJ = sum_h sech^2(pre_h) * c_h with
// c_h = sum_d W2[d,h]*W1[h,d] precomputed once.
//
// Round-5 fix: step-invariant per-lane values (b1, u1, c, b2 slices) hoisted
// into registers ONCE before the step loop -- removes the mid-tile
// s_wait_dscnt 0x0 full drains the compiler was emitting for the sunk bias
// loads, and ~100 ds ops per step. 4-chain GEMM1 retained.

typedef __attribute__((ext_vector_type(2))) float v2f;
typedef __attribute__((ext_vector_type(8))) float v8f;

#define NSTEP 64
#define HSTEP (1.0f / 64.0f)
#define INVL  (1.0f / 0.999f)

// LDS strides (floats), padded against bank conflicts; all v2f offsets even
// so every fragment access is an 8-byte-aligned ds_load_b64.
#define W1ST 66   // W1 stored [256][66]
#define W2ST 258  // W2 stored [64][258]
#define YST  66   // y/z/f tiles stored [16][66]
#define STST 18   // tanh staging tile [16][18]

// LDS layout (float offsets)
static constexpr int OFF_W1 = 0;                      // 256*66
static constexpr int OFF_W2 = OFF_W1 + 256 * W1ST;    // 64*258
static constexpr int OFF_Y  = OFF_W2 + 64 * W2ST;     // 16*66
static constexpr int OFF_Z  = OFF_Y + 16 * YST;
static constexpr int OFF_F  = OFF_Z + 16 * YST;
static constexpr int OFF_ST = OFF_F + 16 * YST;       // 16*18
static constexpr int OFF_B1 = OFF_ST + 16 * STST;     // 256
static constexpr int OFF_U1 = OFF_B1 + 256;           // 256
static constexpr int OFF_B2 = OFF_U1 + 256;           // 64
static constexpr int OFF_C  = OFF_B2 + 64;            // 256
static constexpr int OFF_TS = OFF_C + 256;            // 32*8 trace scratch
static constexpr int OFF_I  = OFF_TS + 256;           // 16
static constexpr int LDS_FLOATS = OFF_I + 16;         // ~148 KB

__device__ __forceinline__ v8f vzero8() {
    v8f z = {0.f, 0.f, 0.f, 0.f, 0.f, 0.f, 0.f, 0.f};
    return z;
}

// Branchless tanh: 1 - 2/(e^{2x}+1), via v_exp_f32 (exp2) and v_rcp_f32.
// Saturates correctly: x->+inf => rcp(inf)=0 => 1; x->-inf => rcp(1)=1 => -1.
__device__ __forceinline__ float fast_tanh(float x) {
    const float LOG2E_X2 = 2.8853900817779268f;  // 2*log2(e)
    float e = __builtin_amdgcn_exp2f(x * LOG2E_X2);
    float r = __builtin_amdgcn_rcpf(e + 1.0f);
    return __builtin_fmaf(-2.0f, r, 1.0f);
}

// One MLP evaluation on the wave's 16-row tile.
//   src: LDS base of the 16x64 input state tile (stride YST)
//   b1v/u1v/ccv/b2v: step-invariant per-lane register slices (see caller)
//   writes f = W2 @ tanh(W1 y + b1 + t u1) + b2 into OFF_F tile
//   if TRACE: accumulates per-lane sech^2 * c into trp[8] (C-layout rows)
template <bool TRACE>
__device__ __forceinline__ void mlp_eval(float* sm, float t, const float* src,
                                         const float* b1v, const float* u1v,
                                         const float* ccv, const float* b2v,
                                         float* trp) {
    const int lane = threadIdx.x & 31;
    const int l16  = lane & 15;   // A: row m / B,C: column n
    const int hi   = lane >> 4;   // lane-group: selects K pair / row+8

    const float* w1 = sm + OFF_W1;
    const float* w2 = sm + OFF_W2;
    float*       st = sm + OFF_ST;
    float*       f  = sm + OFF_F;

    // Hoist the state tile's 16 A-fragments (nt-invariant): 32 VGPRs.
    v2f af[16];
#pragma unroll
    for (int kt = 0; kt < 16; ++kt)
        af[kt] = *(const v2f*)&src[l16 * YST + kt * 4 + 2 * hi];

    v8f facc[4];  // f tile: 16 rows x 64 cols = 4 output N-tiles
#pragma unroll
    for (int ot = 0; ot < 4; ++ot) facc[ot] = vzero8();

#pragma unroll
    for (int nt = 0; nt < 16; ++nt) {  // hidden tiles (H = 256 -> 16x16)
        // ---- batch-issue ALL weight fragments for this N-tile ----
        // GEMM1 B-frags: B[k][n] = W1[n][k]; contiguous K-pair per W1 row.
        v2f bw1[16];
#pragma unroll
        for (int kt = 0; kt < 16; ++kt)
            bw1[kt] = *(const v2f*)&w1[(nt * 16 + l16) * W1ST + kt * 4 + 2 * hi];
        // GEMM2 B-frags: B2[k][n] = W2[n][k]; independent of the staging
        // transpose, so they can load while GEMM1 computes.
        v2f bw2[16];
#pragma unroll
        for (int j = 0; j < 4; ++j)
#pragma unroll
            for (int ot = 0; ot < 4; ++ot)
                bw2[j * 4 + ot] = *(const v2f*)&w2[(ot * 16 + l16) * W2ST +
                                                   nt * 16 + j * 4 + 2 * hi];

        // ---- GEMM1: pre-tile = Y(16x64) x W1^T(64, nt-slice) ----
        // Four independent WMMA chains (kt mod 4) so same-accumulator WMMA
        // latency never stalls issue.
        v8f acc[4];
#pragma unroll
        for (int q = 0; q < 4; ++q) acc[q] = vzero8();
#pragma unroll
        for (int kt = 0; kt < 16; kt += 4) {  // K over D=64, 4 per WMMA
#pragma unroll
            for (int q = 0; q < 4; ++q)
                acc[q] = __builtin_amdgcn_wmma_f32_16x16x4_f32(
                    false, af[kt + q], false, bw1[kt + q], (short)0, acc[q],
                    false, false);
        }
        // tree-merge the four chains (dual-issuable adds)
        v8f pre;
#pragma unroll
        for (int r = 0; r < 8; ++r)
            pre[r] = (acc[0][r] + acc[1][r]) + (acc[2][r] + acc[3][r]);

        // ---- bias + tanh (+ trace): all operands already in registers ----
        const float bias = __builtin_fmaf(t, u1v[nt], b1v[nt]);
        const float cch  = TRACE ? ccv[nt] : 0.f;
#pragma unroll
        for (int r = 0; r < 8; ++r) {  // C element (m = r + 8*hi, n = l16)
            float tv = fast_tanh(pre[r] + bias);
            if (TRACE) trp[r] += (1.0f - tv * tv) * cch;
            // stage in [m][k_local] so it can be reloaded as A-fragments
            st[(r + 8 * hi) * STST + l16] = tv;
        }
        // ---- fused GEMM2 K-chunk: facc += tanh_tile(16x16) x W2^T-slice ----
        // (single wave: LDS ops are in-order, store->load transpose is safe;
        //  4 independent facc chains + resident bw2 give the scheduler ILP)
#pragma unroll
        for (int j = 0; j < 4; ++j) {
            v2f ha = *(const v2f*)&st[l16 * STST + j * 4 + 2 * hi];
#pragma unroll
            for (int ot = 0; ot < 4; ++ot) {
                facc[ot] = __builtin_amdgcn_wmma_f32_16x16x4_f32(
                    false, ha, false, bw2[j * 4 + ot], (short)0, facc[ot],
                    false, false);
            }
        }
    }
    // ---- add b2, scatter f tile to LDS (C-layout, coalesced per VGPR) ----
#pragma unroll
    for (int ot = 0; ot < 4; ++ot) {
#pragma unroll
        for (int r = 0; r < 8; ++r)
            f[(r + 8 * hi) * YST + ot * 16 + l16] = facc[ot][r] + b2v[ot];
    }
}

extern "C" __global__ void __launch_bounds__(32)
cnf_solver_kernel(const float* __restrict__ y1g, const float* __restrict__ W1g,
                  const float* __restrict__ b1g, const float* __restrict__ u1g,
                  const float* __restrict__ W2g, const float* __restrict__ b2g,
                  float* __restrict__ out) {
    extern __shared__ float sm[];
    const int lane = threadIdx.x;      // 32 threads = 1 wave
    const int l16  = lane & 15;
    const int row0 = blockIdx.x * 16;  // this workgroup's batch-row tile

    // Hint the weight streams into cache early (global_prefetch_b8).
    __builtin_prefetch(W1g + lane * 64, 0, 3);
    __builtin_prefetch(W2g + lane * 64, 0, 3);

    // ---- one-time: weights + biases + state into LDS ----
    for (int i2 = lane; i2 < 256 * 32; i2 += 32) {  // W1: 8192 x float2
        int m = i2 >> 5, d = (i2 & 31) * 2;
        *(v2f*)&sm[OFF_W1 + m * W1ST + d] = *(const v2f*)&W1g[m * 64 + d];
    }
    for (int i2 = lane; i2 < 64 * 128; i2 += 32) {  // W2: 8192 x float2
        int m = i2 >> 7, d = (i2 & 127) * 2;
        *(v2f*)&sm[OFF_W2 + m * W2ST + d] = *(const v2f*)&W2g[m * 256 + d];
    }
    for (int i = lane; i < 256; i += 32) {
        sm[OFF_B1 + i] = b1g[i];
        sm[OFF_U1 + i] = u1g[i];
    }
    for (int i = lane; i < 64; i += 32) sm[OFF_B2 + i] = b2g[i];
    for (int i2 = lane; i2 < 16 * 32; i2 += 32) {  // state: 512 x float2
        int m = i2 >> 5, d = (i2 & 31) * 2;
        v2f v = *(const v2f*)&y1g[(row0 + m) * 64 + d];
        *(v2f*)&sm[OFF_Y + m * YST + d] = v;  // y1
        *(v2f*)&sm[OFF_Z + m * YST + d] = v;  // z1 = y1
    }
    if (lane < 16) sm[OFF_I + lane] = 0.f;
    __syncthreads();  // single-wave WG: lowers to nop, kept for form

    // ---- c_h = sum_d W2[d,h] * W1[h,d]  (step-invariant trace weights) ----
    for (int h = lane; h < 256; h += 32) {
        float c = 0.f;
        for (int d = 0; d < 64; ++d)
            c += sm[OFF_W1 + h * W1ST + d] * sm[OFF_W2 + d * W2ST + h];
        sm[OFF_C + h] = c;
    }
    __syncthreads();

    // ---- hoist step-invariant per-lane slices into registers (52 VGPRs) ----
    // Each lane's C-layout column is n = l16, so per N-tile it only ever
    // needs b1/u1/c at index nt*16 + l16, and b2 at ot*16 + l16.
    float b1v[16], u1v[16], ccv[16], b2v[4];
#pragma unroll
    for (int nt = 0; nt < 16; ++nt) {
        b1v[nt] = sm[OFF_B1 + nt * 16 + l16];
        u1v[nt] = sm[OFF_U1 + nt * 16 + l16];
        ccv[nt] = sm[OFF_C + nt * 16 + l16];
    }
#pragma unroll
    for (int ot = 0; ot < 4; ++ot) b2v[ot] = sm[OFF_B2 + ot * 16 + l16];

    // ---- 64 sequential backward Euler steps ----
    float t1 = 1.0f;
#pragma unroll 1
    for (int s = 0; s < NSTEP; ++s) {
        const float t0 = t1 - HSTEP;

        float trp[8];
#pragma unroll
        for (int r = 0; r < 8; ++r) trp[r] = 0.f;

        // f1 = vf(t1, y1), with trace accumulation
        mlp_eval<true>(sm, t1, sm + OFF_Y, b1v, u1v, ccv, b2v, trp);
#pragma unroll
        for (int r = 0; r < 8; r += 2)
            *(v2f*)&sm[OFF_TS + lane * 8 + r] = v2f{trp[r], trp[r + 1]};

        // z0 = z1 - h * f1   (v2f: 16 iters/lane)
        for (int i2 = lane; i2 < 512; i2 += 32) {
            int m = i2 >> 5, d = (i2 & 31) * 2;
            v2f fz = *(const v2f*)&sm[OFF_F + m * YST + d];
            v2f zz = *(const v2f*)&sm[OFF_Z + m * YST + d];
            zz.x -= HSTEP * fz.x;
            zz.y -= HSTEP * fz.y;
            *(v2f*)&sm[OFF_Z + m * YST + d] = zz;
        }

        // I0 = I1 + h * trJ : reduce C-layout partials per row
        if (lane < 16) {
            const int r = lane & 7;           // VGPR slot
            const int g = (lane >> 3) & 1;    // rows 8..15 live in lanes 16..31
            float sum = 0.f;
#pragma unroll
            for (int l = 0; l < 16; ++l) sum += sm[OFF_TS + (l + 16 * g) * 8 + r];
            sm[OFF_I + lane] += HSTEP * sum;
        }

        // f0' = vf(t0, z0)
        mlp_eval<false>(sm, t0, sm + OFF_Z, b1v, u1v, ccv, b2v, trp);

        // y0 = inv_l*y1 + (1-inv_l)*z0 - inv_l*h*f0'   (v2f)
        for (int i2 = lane; i2 < 512; i2 += 32) {
            int m = i2 >> 5, d = (i2 & 31) * 2;
            v2f fz = *(const v2f*)&sm[OFF_F + m * YST + d];
            v2f zz = *(const v2f*)&sm[OFF_Z + m * YST + d];
            v2f yy = *(const v2f*)&sm[OFF_Y + m * YST + d];
            yy.x = INVL * yy.x + (1.0f - INVL) * zz.x - INVL * HSTEP * fz.x;
            yy.y = INVL * yy.y + (1.0f - INVL) * zz.y - INVL * HSTEP * fz.y;
            *(v2f*)&sm[OFF_Y + m * YST + d] = yy;
        }
        t1 = t0;
    }

    // ---- output: [B, 65] = concat(y0, I0) ----
    for (int i = lane; i < 16 * 65; i += 32) {
        int m = i / 65, c = i % 65;
        out[(row0 + m) * 65 + c] = (c < 64) ? sm[OFF_Y + m * YST + c]
                                            : sm[OFF_I + m];
    }
}

extern "C" void kernel_launch(void* const* d_in, const int* in_sizes, int n_in,
                              void* d_out, int out_size, void* d_ws, size_t ws_size,
                              hipStream_t stream) {
    const float* y1 = (const float*)d_in[0];
    const float* W1 = (const float*)d_in[1];
    const float* b1 = (const float*)d_in[2];
    const float* u1 = (const float*)d_in[3];
    const float* W2 = (const float*)d_in[4];
    const float* b2 = (const float*)d_in[5];
    float* out = (float*)d_out;

    const size_t lds_bytes = (size_t)LDS_FLOATS * sizeof(float);  // ~148 KB
    // Allow >64KB dynamic LDS (CDNA5 WGP has 320KB). Deterministic, host-side.
    (void)hipFuncSetAttribute((const void*)cnf_solver_kernel,
                              hipFuncAttributeMaxDynamicSharedMemorySize,
                              (int)lds_bytes);

    // 16 independent row-tiles x 1 wave; each runs all 64 steps.
    cnf_solver_kernel<<<16, 32, lds_bytes, stream>>>(y1, W1, b1, u1, W2, b2, out);
}